// Decoder_35459249995900
// MI455X (gfx1250) — compile-verified
//
#include <hip/hip_runtime.h>

typedef __attribute__((ext_vector_type(2))) float v2f;
typedef __attribute__((ext_vector_type(8))) float v8f;

#define BATCH   262144
#define FEAT    16
#define HID     24
#define NOUT    784
#define NTPW    7      // N-tiles (16 cols each) per wave -> 112 cols
#define NGRP    7      // 49 N-tiles / 7
#define MGRP    2048   // M groups; each covers 8 M-tiles of 16 rows
#define MT_PER  8
#define HSTR    34     // LDS row stride in floats (even -> b64 aligned, conflict-free)
#define WPB     8      // waves per 256-thread block

__device__ __forceinline__ v8f wmma4(v2f a, v2f b, v8f c) {
  // D = A(16x4,f32) * B(4x16,f32) + C(16x16,f32)
  return __builtin_amdgcn_wmma_f32_16x16x4_f32(false, a, false, b, (short)0, c,
                                               false, false);
}

__global__ __launch_bounds__(256) void mlp_fused(
    const float* __restrict__ x, const float* __restrict__ W1,
    const float* __restrict__ b1, const float* __restrict__ W2,
    const float* __restrict__ b2, float* __restrict__ out) {
  __shared__ float hbuf[WPB * 16 * HSTR];

  const int lane = threadIdx.x & 31;
  const int wv   = threadIdx.x >> 5;
  const int gw   = blockIdx.x * WPB + wv;
  const int ng   = gw % NGRP;   // which 112-column slice
  const int mg   = gw / NGRP;   // which 128-row slice
  const int n    = lane & 15;   // N index within tile / M index for A frags
  const int kh   = lane >> 4;   // lane-half

  // ---- resident B fragments of W2 slice (24 x 112) + bias ----
  v2f  bw2[NTPW][6];
  float bias2[NTPW];
#pragma unroll
  for (int t = 0; t < NTPW; ++t) {
    const int col = (ng * NTPW + t) * 16 + n;
    bias2[t] = b2[col];
#pragma unroll
    for (int c = 0; c < 6; ++c) {
      const int k = 4 * c + 2 * kh;          // vgpr j holds K = k + j
      bw2[t][c][0] = W2[k * NOUT + col];
      bw2[t][c][1] = W2[(k + 1) * NOUT + col];
    }
  }

  // ---- resident B fragments of W1 (16 x 24 padded to 16 x 32) + bias ----
  v2f  bw1[2][4];
  float bias1[2];
#pragma unroll
  for (int t = 0; t < 2; ++t) {
    const int col = t * 16 + n;
    const bool ok = (col < HID);
    bias1[t] = ok ? b1[col] : 0.0f;
#pragma unroll
    for (int c = 0; c < 4; ++c) {
      const int k = 4 * c + 2 * kh;
      bw1[t][c][0] = ok ? W1[k * HID + col] : 0.0f;
      bw1[t][c][1] = ok ? W1[(k + 1) * HID + col] : 0.0f;
    }
  }

  float* hs = hbuf + wv * 16 * HSTR;  // wave-private h staging

  for (int it = 0; it < MT_PER; ++it) {
    const int mbase = mg * (MT_PER * 16) + it * 16;

    // A fragments of x tile (16 x 16): lane holds x[m=n][2kh + j] per chunk
    const float* xp = x + (size_t)(mbase + n) * FEAT + 2 * kh;
    v2f xa[4];
#pragma unroll
    for (int c = 0; c < 4; ++c) xa[c] = *(const v2f*)(xp + 4 * c);

    // ---- GEMM1: h = relu(x @ W1 + b1), write to LDS in row-major ----
#pragma unroll
    for (int t = 0; t < 2; ++t) {
      v8f acc;
#pragma unroll
      for (int r = 0; r < 8; ++r) acc[r] = bias1[t];
#pragma unroll
      for (int c = 0; c < 4; ++c) acc = wmma4(xa[c], bw1[t][c], acc);
      // D layout: vgpr r, lane -> h[r + 8*kh][n]
#pragma unroll
      for (int r = 0; r < 8; ++r)
        hs[(r + 8 * kh) * HSTR + t * 16 + n] = fmaxf(acc[r], 0.0f);
    }

    // LDS ops are in-order per wave; just stop the compiler from reordering.
    asm volatile("" ::: "memory");

    // A fragments of h (16 x 24): ds_load_b64 per chunk
    const float* hl = hs + n * HSTR + 2 * kh;
    v2f ha[6];
#pragma unroll
    for (int c = 0; c < 6; ++c) ha[c] = *(const v2f*)(hl + 4 * c);

    // ---- GEMM2 + clamp(0.25*z + 0.5, 0, 1) + store ----
    float* op = out + (size_t)(mbase + 8 * kh) * NOUT + ng * (NTPW * 16) + n;
#pragma unroll
    for (int t = 0; t < NTPW; ++t) {
      v8f acc;
#pragma unroll
      for (int r = 0; r < 8; ++r) acc[r] = bias2[t];
#pragma unroll
      for (int c = 0; c < 6; ++c) acc = wmma4(ha[c], bw2[t][c], acc);
#pragma unroll
      for (int r = 0; r < 8; ++r) {
        float v = fmaf(acc[r], 0.25f, 0.5f);
        v = fminf(fmaxf(v, 0.0f), 1.0f);
        op[(size_t)r * NOUT + t * 16] = v;
      }
    }
  }
}

extern "C" void kernel_launch(void* const* d_in, const int* in_sizes, int n_in,
                              void* d_out, int out_size, void* d_ws, size_t ws_size,
                              hipStream_t stream) {
  const float* x  = (const float*)d_in[0];
  const float* W1 = (const float*)d_in[1];
  const float* b1 = (const float*)d_in[2];
  const float* W2 = (const float*)d_in[3];
  const float* b2 = (const float*)d_in[4];
  float* out = (float*)d_out;

  const int total_waves = NGRP * MGRP;       // 14336
  const int blocks = total_waves / WPB;      // 1792
  mlp_fused<<<blocks, 256, 0, stream>>>(x, W1, b1, W2, b2, out);
}